// GraphConvBlock_35416300323760
// MI455X (gfx1250) — compile-verified
//
#include <hip/hip_runtime.h>

// Problem constants (match reference)
#define N_NODES   100000
#define N_EDGES   1600000
#define DCH       64          // D_IN == D_OUT
#define NGRAPH    64
#define GN_EPS    1e-5f

typedef float v2f __attribute__((ext_vector_type(2)));
typedef float v8f __attribute__((ext_vector_type(8)));

// ---------------------------------------------------------------------------
// Workspace layout (floats):
//   [0]                 agg      N*64
//   [OFF_GSUM]          gsum     64*64
//   [OFF_GSQ]           gsumsq   64*64
//   [OFF_GCNT]          gcnt     64
//   [OFF_H]             h        N*64
//   [OFF_GSCALE]        gscale   64*64
//   [OFF_GSHIFT]        gshift   64*64
// agg..gcnt are contiguous so a single zero pass covers everything that
// needs zero-init each call (harness does not re-poison between replays).
// ---------------------------------------------------------------------------
#define OFF_AGG    0
#define OFF_GSUM   (N_NODES * DCH)
#define OFF_GSQ    (OFF_GSUM + NGRAPH * DCH)
#define OFF_GCNT   (OFF_GSQ + NGRAPH * DCH)
#define OFF_H      (OFF_GCNT + NGRAPH)
#define OFF_GSCALE (OFF_H + N_NODES * DCH)
#define OFF_GSHIFT (OFF_GSCALE + NGRAPH * DCH)
#define ZERO_COUNT (OFF_H)   // zero agg + gsum + gsumsq + gcnt

// ---------------------------------------------------------------------------
__global__ void zero_kernel(float* __restrict__ p, long long n) {
    long long i = (long long)blockIdx.x * blockDim.x + threadIdx.x;
    long long stride = (long long)gridDim.x * blockDim.x;
    for (; i < n; i += stride) p[i] = 0.0f;
}

// One thread per (edge, channel): msg = edge_attr[e] * x[src[e], d] scattered
// into agg[dst[e], d]. Within a wave, lanes cover consecutive d of the same
// edge -> coalesced 128B x reads; agg updates via global_atomic_add_f32.
__global__ void edge_scatter_kernel(const float* __restrict__ x,
                                    const int*   __restrict__ eidx,
                                    const float* __restrict__ eattr,
                                    float*       __restrict__ agg) {
    long long idx = (long long)blockIdx.x * blockDim.x + threadIdx.x;
    if (idx >= (long long)N_EDGES * DCH) return;
    int e = (int)(idx >> 6);
    int d = (int)(idx & 63);
    int s  = eidx[e];              // row 0 of edge_index
    int dn = eidx[N_EDGES + e];    // row 1 of edge_index
    float v = eattr[e] * x[(long long)s * DCH + d];
    atomicAdd(&agg[(long long)dn * DCH + d], v);
}

// Per-graph node counts (float, used as divisor later).
__global__ void count_nodes_kernel(const int* __restrict__ batch,
                                   float* __restrict__ gcnt) {
    int n = blockIdx.x * blockDim.x + threadIdx.x;
    if (n < N_NODES) atomicAdd(&gcnt[batch[n]], 1.0f);
}

// ---------------------------------------------------------------------------
// Fused: h = relu(agg @ W_rel^T + x @ W_root^T + b_rel), plus GraphNorm
// sum / sum-of-squares accumulation straight from the WMMA accumulators.
// One wave32 computes a 16-row x 64-col tile of h using V_WMMA_F32_16X16X4_F32.
//
// Weights are staged in LDS in pair-interleaved "fragment" layout so that a
// full B fragment (two K-adjacent values for one output column) is a single
// aligned ds_load_b64 directly into an even VGPR pair:
//   Wlds[(k2*64 + o)*2 + j] = W[o][2*k2 + j],  j in {0,1}
// ---------------------------------------------------------------------------
__launch_bounds__(128, 1)
__global__ void gemm_wmma_kernel(const float* __restrict__ agg,
                                 const float* __restrict__ x,
                                 const float* __restrict__ Wrel,
                                 const float* __restrict__ brel,
                                 const float* __restrict__ Wroot,
                                 const int*   __restrict__ batch,
                                 float* __restrict__ h,
                                 float* __restrict__ gsum,
                                 float* __restrict__ gsumsq) {
    __shared__ v2f WldsRel[DCH * DCH / 2];   // indexed by k2*64 + o
    __shared__ v2f WldsRoot[DCH * DCH / 2];

    // Stage: thread i handles pair (o = i>>5, k2 = i&31) -> coalesced b64
    // global reads (consecutive k2 across lanes = consecutive addresses).
    for (int i = threadIdx.x; i < DCH * DCH / 2; i += blockDim.x) {
        int o = i >> 5, k2 = i & 31;
        WldsRel[k2 * DCH + o]  = *(const v2f*)(Wrel  + o * DCH + 2 * k2);
        WldsRoot[k2 * DCH + o] = *(const v2f*)(Wroot + o * DCH + 2 * k2);
    }
    __syncthreads();

    const int wave = threadIdx.x >> 5;
    const int lane = threadIdx.x & 31;
    const int tile = blockIdx.x * 4 + wave;      // 16-row tile index
    const int numTiles = N_NODES / 16;           // 6250 (exact)
    if (tile >= numTiles) return;                // wave-uniform guard

    const int r0    = tile * 16;
    const int mrow  = lane & 15;                 // M (A) / N (B,C,D) position
    const int khalf = lane >> 4;                 // lanes 16-31 own K+2 / M+8

    // Per-lane base pointers into the fragment-major LDS arrays:
    // fragment (s, t) sits at bp[s*128 + t*16]  (k2 = 2s + khalf, o = t*16+mrow)
    const v2f* bpRel  = &WldsRel[khalf * DCH + mrow];
    const v2f* bpRoot = &WldsRoot[khalf * DCH + mrow];

    v8f acc[4] = {v8f{0}, v8f{0}, v8f{0}, v8f{0}};

    // ---- GEMM 1: agg @ W_rel^T ------------------------------------------
    {
        const float* arow = agg + (long long)(r0 + mrow) * DCH + 2 * khalf;
        #pragma unroll
        for (int s = 0; s < 16; ++s) {
            v2f a = *(const v2f*)(arow + 4 * s);   // A[M][kb+2kh .. +1]
            #pragma unroll
            for (int t = 0; t < 4; ++t) {
                v2f b = bpRel[s * 128 + t * 16];   // single ds_load_b64
                acc[t] = __builtin_amdgcn_wmma_f32_16x16x4_f32(
                    false, a, false, b, (short)0, acc[t], false, false);
            }
        }
    }
    // ---- GEMM 2: x @ W_root^T (same accumulators) -----------------------
    {
        const float* arow = x + (long long)(r0 + mrow) * DCH + 2 * khalf;
        #pragma unroll
        for (int s = 0; s < 16; ++s) {
            v2f a = *(const v2f*)(arow + 4 * s);
            #pragma unroll
            for (int t = 0; t < 4; ++t) {
                v2f b = bpRoot[s * 128 + t * 16];
                acc[t] = __builtin_amdgcn_wmma_f32_16x16x4_f32(
                    false, a, false, b, (short)0, acc[t], false, false);
            }
        }
    }

    // ---- bias + relu + store h + GraphNorm stats ------------------------
    // C/D layout: VGPR v, lanes 0-15 -> h[r0+v][o], lanes 16-31 -> h[r0+v+8][o]
    const int rowoff = khalf * 8;
    #pragma unroll
    for (int t = 0; t < 4; ++t) {
        const int o = t * 16 + mrow;
        const float bias = brel[o];
        #pragma unroll
        for (int v = 0; v < 8; ++v) {
            float val = acc[t][v] + bias;
            val = val > 0.0f ? val : 0.0f;
            const int row = r0 + v + rowoff;
            h[(long long)row * DCH + o] = val;
            const int g = batch[row];
            atomicAdd(&gsum[g * DCH + o], val);
            atomicAdd(&gsumsq[g * DCH + o], val * val);
        }
    }
}

// ---------------------------------------------------------------------------
// Per-(graph,channel) stats -> affine scale/shift.
// out  = h - mean*s ;  var = E[h^2] - (2s - s^2)*mean^2
// y = gw*(h - mean*s)/sqrt(var+eps) + gb = scale*h + shift
// ---------------------------------------------------------------------------
__global__ void finalize_stats_kernel(const float* __restrict__ gsum,
                                      const float* __restrict__ gsumsq,
                                      const float* __restrict__ gcnt,
                                      const float* __restrict__ gw,
                                      const float* __restrict__ gb,
                                      const float* __restrict__ gms,
                                      float* __restrict__ gscale,
                                      float* __restrict__ gshift) {
    int i = blockIdx.x * blockDim.x + threadIdx.x;
    if (i >= NGRAPH * DCH) return;
    int g = i >> 6, d = i & 63;
    float cnt  = gcnt[g];
    float inv  = cnt > 0.0f ? 1.0f / cnt : 0.0f;
    float mean = gsum[i] * inv;
    float msq  = gsumsq[i] * inv;
    float s    = gms[d];
    float var  = msq - (2.0f * s - s * s) * mean * mean;
    float a    = gw[d] * rsqrtf(var + GN_EPS);
    gscale[i] = a;
    gshift[i] = gb[d] - a * (mean * s);
}

// Apply per-(graph,channel) affine; float4 vectorized (global_load_b128).
__global__ void normalize_kernel(const float* __restrict__ h,
                                 const int*   __restrict__ batch,
                                 const float* __restrict__ gscale,
                                 const float* __restrict__ gshift,
                                 float* __restrict__ out) {
    long long idx = (long long)blockIdx.x * blockDim.x + threadIdx.x;
    if (idx >= (long long)N_NODES * (DCH / 4)) return;
    int n  = (int)(idx >> 4);
    int d4 = (int)(idx & 15) * 4;
    int g  = batch[n];
    const float4 hv = *(const float4*)(h + (long long)n * DCH + d4);
    const float4 sc = *(const float4*)(gscale + g * DCH + d4);
    const float4 sh = *(const float4*)(gshift + g * DCH + d4);
    float4 y;
    y.x = sc.x * hv.x + sh.x;
    y.y = sc.y * hv.y + sh.y;
    y.z = sc.z * hv.z + sh.z;
    y.w = sc.w * hv.w + sh.w;
    *(float4*)(out + (long long)n * DCH + d4) = y;
}

// ---------------------------------------------------------------------------
extern "C" void kernel_launch(void* const* d_in, const int* in_sizes, int n_in,
                              void* d_out, int out_size, void* d_ws, size_t ws_size,
                              hipStream_t stream) {
    const float* x      = (const float*)d_in[0];
    const int*   eidx   = (const int*)d_in[1];
    const float* eattr  = (const float*)d_in[2];
    const int*   batch  = (const int*)d_in[3];
    const float* Wrel   = (const float*)d_in[4];
    const float* brel   = (const float*)d_in[5];
    const float* Wroot  = (const float*)d_in[6];
    const float* gw     = (const float*)d_in[7];
    const float* gb     = (const float*)d_in[8];
    const float* gms    = (const float*)d_in[9];
    float* out = (float*)d_out;

    float* ws     = (float*)d_ws;
    float* agg    = ws + OFF_AGG;
    float* gsum   = ws + OFF_GSUM;
    float* gsumsq = ws + OFF_GSQ;
    float* gcnt   = ws + OFF_GCNT;
    float* h      = ws + OFF_H;
    float* gscale = ws + OFF_GSCALE;
    float* gshift = ws + OFF_GSHIFT;

    // 1) zero agg + stats (contiguous region)
    zero_kernel<<<2048, 256, 0, stream>>>(ws, (long long)ZERO_COUNT);

    // 2) edge-weighted gather/scatter into agg
    {
        long long total = (long long)N_EDGES * DCH;
        int blocks = (int)((total + 255) / 256);
        edge_scatter_kernel<<<blocks, 256, 0, stream>>>(x, eidx, eattr, agg);
    }

    // 3) per-graph node counts
    count_nodes_kernel<<<(N_NODES + 255) / 256, 256, 0, stream>>>(batch, gcnt);

    // 4) fused dual-GEMM + bias + relu + stats (WMMA f32 16x16x4)
    {
        int numTiles = N_NODES / 16;                 // 6250
        int blocks = (numTiles + 3) / 4;             // 4 waves / block
        gemm_wmma_kernel<<<blocks, 128, 0, stream>>>(agg, x, Wrel, brel, Wroot,
                                                     batch, h, gsum, gsumsq);
    }

    // 5) stats -> affine
    finalize_stats_kernel<<<(NGRAPH * DCH + 255) / 256, 256, 0, stream>>>(
        gsum, gsumsq, gcnt, gw, gb, gms, gscale, gshift);

    // 6) apply normalization
    {
        long long total = (long long)N_NODES * (DCH / 4);
        int blocks = (int)((total + 255) / 256);
        normalize_kernel<<<blocks, 256, 0, stream>>>(h, batch, gscale, gshift, out);
    }
}